// QuantizedAdd_66408784331401
// MI455X (gfx1250) — compile-verified
//
#include <hip/hip_runtime.h>

// Native vector types so __builtin_nontemporal_load/store get real 128-bit accesses.
typedef int   v4i __attribute__((ext_vector_type(4)));
typedef float v4f __attribute__((ext_vector_type(4)));

namespace {
constexpr int kB              = 128;
constexpr int kCHW            = 64 * 56 * 56;               // 200704 elements per sample
constexpr int kThreads        = 256;                        // 8 wave32 waves per block
constexpr int kElemsPerThread = 4;                          // one B128 per array per thread
constexpr int kElemsPerBlock  = kThreads * kElemsPerThread; // 1024
constexpr int kBlocksPerSample= kCHW / kElemsPerBlock;      // 196, exact -> block is sample-uniform
static_assert(kBlocksPerSample * kElemsPerBlock == kCHW, "block must not straddle samples");
}

// Per-tensor requantize parameters, all block-uniform (SGPR-resident).
// Identities baked in (proofs in commit message / analysis):
//   ((v<<lsh)*M0 + nudge) >> 31  ==  (v*M0 + nd) >> (31-lsh)
//     nd = v>=0 ? 2^(30-lsh) : (lsh==0 ? 1-2^30 : -2^(30-lsh))
//   RoundingDivideByPOT(r, s)    ==  (r + (s? 2^(s-1) - (r<0) : 0)) >> s
//   sign(prod) == sign(r) == sign(v)   (M0 >= 2^30 > 0; |v|>=1 => |r|>=1)
struct RqParams { int z, m0, ndP, ndN, sh31, biasP, biasN, rsh; };

__device__ __forceinline__ RqParams make_params(int z, int m0, int s) {
    RqParams p;
    p.z  = z;
    p.m0 = m0;
    const int lsh = (s < 0) ? -s : 0;       // lsh in {0,1,2}
    p.rsh   = (s < 0) ? 0 : s;              // rsh in [0,9]
    p.sh31  = 31 - lsh;
    p.ndP   = 1 << (30 - lsh);
    p.ndN   = (lsh == 0) ? (1 - (1 << 30)) : -(1 << (30 - lsh));
    p.biasP = (p.rsh > 0) ? (1 << (p.rsh - 1)) : 0;
    p.biasN = (p.rsh > 0) ? (p.biasP - 1)      : 0;
    return p;
}

// Exact gemmlowp SRDHM + RoundingDivideByPOT for x,z in [0,255].
// VALU shape: sub, cmp(v<0) shared by two cndmasks, mul_lo+mul_hi, 64b add,
// 64b shift/extract, add, ashr.
__device__ __forceinline__ int requant(int x, const RqParams& p) {
    const int       v    = x - p.z;                         // |v| <= 255
    const bool      neg  = v < 0;                           // == sign(prod) == sign(r)
    const long long prod = (long long)v * (long long)p.m0;  // fits in 41 bits
    const long long nd   = (long long)(neg ? p.ndN : p.ndP);
    const int       r    = (int)((prod + nd) >> p.sh31);    // |r| <= 1020
    const int       bias = neg ? p.biasN : p.biasP;
    return (r + bias) >> p.rsh;
}

__global__ __launch_bounds__(kThreads) void QuantizedAdd_stream_kernel(
    const int* __restrict__ bypass,       const int* __restrict__ prev,
    const int* __restrict__ batch_cluster,
    const int* __restrict__ z_bypass,     const int* __restrict__ z_prev,
    const int* __restrict__ z3,
    const int* __restrict__ M0_bypass,    const int* __restrict__ M0_prev,
    const int* __restrict__ shift_bypass, const int* __restrict__ shift_prev,
    float* __restrict__ out)
{
    const unsigned blk = blockIdx.x;
    const int      b   = (int)(blk / kBlocksPerSample);     // uniform -> SALU magic-div

    // Block-uniform parameter setup: scalar loads + SALU only.
    const int c = batch_cluster[b];
    const RqParams pb = make_params(z_bypass[c], M0_bypass[c], shift_bypass[c]);
    const RqParams pp = make_params(z_prev[c],   M0_prev[c],   shift_prev[c]);
    const int z3v = z3[c];

    // Samples are contiguous, so the element offset is just linear over the
    // whole tensor. 32-bit offset (byte offset < 2^31) -> SADDR+VGPR32 global form.
    const unsigned base = blk * (unsigned)kElemsPerBlock
                        + (unsigned)threadIdx.x * kElemsPerThread;  // 16B-aligned

    // One-pass 308MB stream (> 192MB L2): non-temporal B128 loads/stores.
    const v4i xb = __builtin_nontemporal_load((const v4i*)(bypass + base));
    const v4i xp = __builtin_nontemporal_load((const v4i*)(prev   + base));

    v4f res;
#pragma unroll
    for (int i = 0; i < kElemsPerThread; ++i) {
        const int r1 = requant(xb[i], pb);
        const int r2 = requant(xp[i], pp);
        int t = r1 + r2 + z3v;                  // v_add3_u32
        t = t < -128 ? -128 : (t > 127 ? 127 : t);  // v_med3_i32
        res[i] = (float)t;                      // v_cvt_f32_i32
    }
    __builtin_nontemporal_store(res, (v4f*)(out + base));
}

extern "C" void kernel_launch(void* const* d_in, const int* in_sizes, int n_in,
                              void* d_out, int out_size, void* d_ws, size_t ws_size,
                              hipStream_t stream) {
    (void)in_sizes; (void)n_in; (void)out_size; (void)d_ws; (void)ws_size;

    const int* bypass        = (const int*)d_in[0];
    const int* prev          = (const int*)d_in[1];
    const int* batch_cluster = (const int*)d_in[2];
    const int* z_bypass      = (const int*)d_in[3];
    const int* z_prev        = (const int*)d_in[4];
    const int* z3            = (const int*)d_in[5];
    const int* M0_bypass     = (const int*)d_in[6];
    const int* M0_prev       = (const int*)d_in[7];
    const int* shift_bypass  = (const int*)d_in[8];
    const int* shift_prev    = (const int*)d_in[9];
    float*     out           = (float*)d_out;

    const dim3 grid(kB * kBlocksPerSample);   // 128 * 196 = 25088 blocks
    const dim3 block(kThreads);
    QuantizedAdd_stream_kernel<<<grid, block, 0, stream>>>(
        bypass, prev, batch_cluster, z_bypass, z_prev, z3,
        M0_bypass, M0_prev, shift_bypass, shift_prev, out);
}